// RobertaCRFForNER_38156489458437
// MI455X (gfx1250) — compile-verified
//
#include <hip/hip_runtime.h>

typedef __attribute__((ext_vector_type(2))) float v2f;
typedef __attribute__((ext_vector_type(8))) float v8f;

constexpr int kB  = 8192;
constexpr int kT  = 512;
constexpr int kNT = 7;
constexpr int kRowsPerWave   = 16;
constexpr int kWavesPerBlock = 4;

__device__ __forceinline__ int clampTag(int t) {
  t = t < 0 ? 0 : t;
  return t > (kNT - 1) ? (kNT - 1) : t;
}

// 7-way select (v_cmp/v_cndmask chain, no divergence)
__device__ __forceinline__ float sel7(int k, float e0, float e1, float e2,
                                      float e3, float e4, float e5, float e6) {
  float v = e0;
  v = (k == 1) ? e1 : v;
  v = (k == 2) ? e2 : v;
  v = (k == 3) ? e3 : v;
  v = (k == 4) ? e4 : v;
  v = (k == 5) ? e5 : v;
  v = (k == 6) ? e6 : v;
  return v;
}

__device__ __forceinline__ float swapHalves(float x) {
  // lane n <-> lane n^16 (group-of-32 swizzle, xor_mask=0x10, and_mask=0x1f)
  return __int_as_float(__builtin_amdgcn_ds_swizzle(__float_as_int(x), 0x401F));
}

// Forward CRF scan in scaled-probability space.
//
// Each wave32 owns 16 batch rows. alpha lives in 7 VGPRs c0..c6 with a
// *dual-state* layout chosen so the WMMA D operand of step t is directly the
// WMMA B operand of step t+1 (no cross-lane traffic in the loop):
//   c_i lanes 0-15  = state i     of batch row n (= lane)
//   c_i lanes 16-31 = state i+2   of batch row n (= lane-16)   [i+2>6 -> 0]
// A (16x4 f32) holds ExpT^T with rows M=0..7 -> states 0..7 and rows
// M=8..15 -> states 2..9 (states >= 7 are zero padding), so D's upper-half
// rows reproduce states 2..9 in exactly the B-layout positions.
__global__ __launch_bounds__(kWavesPerBlock * 32)
void crf_fwd_kernel(const float* __restrict__ emis,
                    const int* __restrict__ labels,
                    const int* __restrict__ mask,
                    const float* __restrict__ startT,
                    const float* __restrict__ endT,
                    const float* __restrict__ trans,
                    float* __restrict__ partials) {
  __shared__ float sTrans[kNT * kNT];
  __shared__ float sStart[kNT];
  __shared__ float sEnd[kNT];
  {
    int tid = threadIdx.x;
    if (tid < kNT * kNT) sTrans[tid] = trans[tid];
    if (tid >= 64 && tid < 64 + kNT) sStart[tid - 64] = startT[tid - 64];
    if (tid >= 96 && tid < 96 + kNT) sEnd[tid - 96] = endT[tid - 96];
  }
  __syncthreads();

  const int lane = threadIdx.x & 31;
  const int wave = threadIdx.x >> 5;
  const int waveGlobal = blockIdx.x * kWavesPerBlock + wave;
  const int m    = lane & 15;      // A-matrix M row index for this lane
  const int half = lane >> 4;
  const bool lowHalf = (lane < 16);

  // A[M][K] = ET(K, state(M)), state(M) = M<8 ? M : M-6  (rows for states 2..9)
  v2f a0, a1;
  {
    const int j = (m < 8) ? m : (m - 6);
    auto ET = [&](int k, int jj) -> float {
      return (k < kNT && jj < kNT) ? __expf(sTrans[k * kNT + jj]) : 0.0f;
    };
    a0.x = ET(2 * half + 0, j);
    a0.y = ET(2 * half + 1, j);
    a1.x = ET(4 + 2 * half + 0, j);
    a1.y = ET(4 + 2 * half + 1, j);
  }

  const int row = waveGlobal * kRowsPerWave + m;   // lanes 16-31 duplicate a row
  const float* ePtr = emis + (size_t)row * kT * kNT;
  const int*   lPtr = labels + (size_t)row * kT;
  const int*   mPtr = mask   + (size_t)row * kT;

  // ---- t = 0: alpha = exp(start + emit0) in dual-state layout ----
  float e0 = ePtr[0], e1 = ePtr[1], e2 = ePtr[2], e3 = ePtr[3],
        e4 = ePtr[4], e5 = ePtr[5], e6 = ePtr[6];
  float c0 = __expf((lowHalf ? sStart[0] : sStart[2]) + (lowHalf ? e0 : e2));
  float c1 = __expf((lowHalf ? sStart[1] : sStart[3]) + (lowHalf ? e1 : e3));
  float c2 = __expf((lowHalf ? sStart[2] : sStart[4]) + (lowHalf ? e2 : e4));
  float c3 = __expf((lowHalf ? sStart[3] : sStart[5]) + (lowHalf ? e3 : e5));
  float c4 = __expf((lowHalf ? sStart[4] : sStart[6]) + (lowHalf ? e4 : e6));
  float c5 = lowHalf ? __expf(sStart[5] + e5) : 0.0f;   // upper = state 7 = 0
  float c6 = lowHalf ? __expf(sStart[6] + e6) : 0.0f;   // upper = state 8 = 0

  int prev = clampTag(lPtr[0]);
  float num = sStart[prev] + sel7(prev, e0, e1, e2, e3, e4, e5, e6);
  int maskSum = mPtr[0];
  float logZ = 0.0f;

  // One scan step given this step's 7 raw emissions + label + mask.
  auto stepV = [&](float f0, float f1, float f2, float f3, float f4, float f5,
                   float f6, int lab, int mt) {
    const int cur = clampTag(lab);

    // B of this step IS the c registers (D layout of previous step).
    v2f blo, bhi;
    blo.x = c0; blo.y = c1;
    bhi.x = c4; bhi.y = c5;
    v8f acc = {0.0f, 0.0f, 0.0f, 0.0f, 0.0f, 0.0f, 0.0f, 0.0f};
    acc = __builtin_amdgcn_wmma_f32_16x16x4_f32(false, a0, false, blo,
                                                (short)0, acc, false, false);
    acc = __builtin_amdgcn_wmma_f32_16x16x4_f32(false, a1, false, bhi,
                                                (short)0, acc, false, false);

    // Elementwise * exp(emit): upper half needs state i+2's emission.
    const float x0 = __expf(lowHalf ? f0 : f2);
    const float x1 = __expf(lowHalf ? f1 : f3);
    const float x2 = __expf(lowHalf ? f2 : f4);
    const float x3 = __expf(lowHalf ? f3 : f5);
    const float x4 = __expf(lowHalf ? f4 : f6);
    const float x5 = __expf(f5);   // upper multiplies 0 (state 7)
    const float x6 = __expf(f6);   // upper multiplies 0 (state 8)

    const bool keep = (mt > 0);
    c0 = keep ? acc[0] * x0 : c0;
    c1 = keep ? acc[1] * x1 : c1;
    c2 = keep ? acc[2] * x2 : c2;
    c3 = keep ? acc[3] * x3 : c3;
    c4 = keep ? acc[4] * x4 : c4;
    c5 = keep ? acc[5] * x5 : c5;
    c6 = keep ? acc[6] * x6 : c6;

    // ---- numerator (rows on lanes 0-15) ----
    const float stepScore =
        sTrans[prev * kNT + cur] + sel7(cur, f0, f1, f2, f3, f4, f5, f6);
    num += (float)mt * stepScore;
    maskSum += mt;
    prev = cur;
  };

  auto renorm = [&]() {
    // True sum only valid on lanes 0-15 (states 0..6); broadcast to upper half
    // so the duplicated states stay consistent.
    float s = ((c0 + c1) + (c2 + c3)) + ((c4 + c5) + c6);
    const float sw = swapHalves(s);
    s = lowHalf ? s : sw;
    s = fmaxf(s, 1e-30f);
    logZ += __logf(s);
    const float inv = __builtin_amdgcn_rcpf(s);
    c0 *= inv; c1 *= inv; c2 *= inv; c3 *= inv;
    c4 *= inv; c5 *= inv; c6 *= inv;
  };

  // ---- prologue: steps 1..3 (scalar emission loads) ----
  for (int t = 1; t <= 3; ++t) {
    const float* ep = ePtr + t * kNT;
    stepV(ep[0], ep[1], ep[2], ep[3], ep[4], ep[5], ep[6], lPtr[t], mPtr[t]);
  }
  renorm();

  // ---- main loop: chunks of 4 steps. At t%4==0 the 28 emission floats are
  // contiguous and 16B-aligned (row stride 14336B, 4*7*4=112B per chunk), so
  // fetch each chunk as 7x global_load_b128 + int4 labels/mask. ----
  for (int t = 4; t < kT; t += 4) {
    __builtin_prefetch(ePtr + (size_t)(t + 16 < kT ? t + 16 : t) * kNT, 0, 1);
    const float4* ev = reinterpret_cast<const float4*>(ePtr + (size_t)t * kNT);
    const float4 qa = ev[0], qb = ev[1], qc = ev[2], qd = ev[3],
                 qe = ev[4], qf = ev[5], qg = ev[6];
    const int4 lv = *reinterpret_cast<const int4*>(lPtr + t);
    const int4 mv = *reinterpret_cast<const int4*>(mPtr + t);

    stepV(qa.x, qa.y, qa.z, qa.w, qb.x, qb.y, qb.z, lv.x, mv.x);   // t+0
    stepV(qb.w, qc.x, qc.y, qc.z, qc.w, qd.x, qd.y, lv.y, mv.y);   // t+1
    stepV(qd.z, qd.w, qe.x, qe.y, qe.z, qe.w, qf.x, lv.z, mv.z);   // t+2
    stepV(qf.y, qf.z, qf.w, qg.x, qg.y, qg.z, qg.w, lv.w, mv.w);   // t+3
    renorm();
  }

  // ---- den = logZ + log(sum_j alpha_j * exp(end_j)) (lanes 0-15 valid) ----
  const float z = c0 * __expf(sEnd[0]) + c1 * __expf(sEnd[1]) +
                  c2 * __expf(sEnd[2]) + c3 * __expf(sEnd[3]) +
                  c4 * __expf(sEnd[4]) + c5 * __expf(sEnd[5]) +
                  c6 * __expf(sEnd[6]);
  const float den = logZ + __logf(fmaxf(z, 1e-30f));

  int seqEnd = maskSum - 1;
  seqEnd = seqEnd < 0 ? 0 : (seqEnd > kT - 1 ? kT - 1 : seqEnd);
  num += sEnd[clampTag(lPtr[seqEnd])];

  float val = lowHalf ? (den - num) : 0.0f;
#pragma unroll
  for (int off = 16; off > 0; off >>= 1) val += __shfl_xor(val, off, 32);
  if (lane == 0) partials[waveGlobal] = val;
}

// Deterministic fixed-order reduction of the 512 per-wave partials.
__global__ __launch_bounds__(512)
void crf_reduce_kernel(const float* __restrict__ partials,
                       float* __restrict__ out, int n) {
  __shared__ float s[512];
  const int tid = threadIdx.x;
  s[tid] = (tid < n) ? partials[tid] : 0.0f;
  __syncthreads();
#pragma unroll
  for (int off = 256; off > 0; off >>= 1) {
    if (tid < off) s[tid] += s[tid + off];
    __syncthreads();
  }
  if (tid == 0) out[0] = s[0];
}

extern "C" void kernel_launch(void* const* d_in, const int* in_sizes, int n_in,
                              void* d_out, int out_size, void* d_ws, size_t ws_size,
                              hipStream_t stream) {
  const float* emis   = (const float*)d_in[0];
  const int*   labels = (const int*)d_in[1];
  const int*   mask   = (const int*)d_in[2];
  const float* startT = (const float*)d_in[3];
  const float* endT   = (const float*)d_in[4];
  const float* trans  = (const float*)d_in[5];
  float* out      = (float*)d_out;
  float* partials = (float*)d_ws;   // 512 floats

  const int numWaves  = kB / kRowsPerWave;              // 512
  const int numBlocks = numWaves / kWavesPerBlock;      // 128

  crf_fwd_kernel<<<numBlocks, kWavesPerBlock * 32, 0, stream>>>(
      emis, labels, mask, startT, endT, trans, partials);
  crf_reduce_kernel<<<1, 512, 0, stream>>>(partials, out, numWaves);
}